// GCN_42657615184589
// MI455X (gfx1250) — compile-verified
//
#include <hip/hip_runtime.h>
#include <hip/hip_bf16.h>

// ---------------------------------------------------------------------------
// 2-layer GCN on gfx1250 (MI455X).
//  - GEMM via V_WMMA_F32_16X16X4_F32 (native fp32 matrix op, matches ref dtype)
//  - W-tile staged via GLOBAL_LOAD_ASYNC_TO_LDS_B128 (ASYNCcnt path, CDNA5)
//  - Edge aggregation: wave-per-edge, float4 gather + 4x global_atomic_add_f32
//    (h / out buffers are 25.6MB each -> L2 resident; atomic-bound, not HBM)
// ---------------------------------------------------------------------------

typedef float v2f __attribute__((ext_vector_type(2)));
typedef float v8f __attribute__((ext_vector_type(8)));

#define DF   128          // feature dim (all layers)
#define LDSP 132          // padded LDS row stride (floats): 132%64=4 -> no bank conflicts
#define KC   64           // K-chunk of W staged in LDS per phase

// LDS byte offset of a __shared__ object (generic -> addrspace(3) cast)
__device__ __forceinline__ unsigned lds_offset_of(const void* p) {
    return (unsigned)(size_t)(const __attribute__((address_space(3))) void*)p;
}

// ---------------- degree / normalization ----------------

__global__ void zero_k(float* p, int n) {
    int i = blockIdx.x * blockDim.x + threadIdx.x;
    if (i < n) p[i] = 0.0f;
}

__global__ void count_deg_k(const int* __restrict__ dst, float* deg, int nE) {
    int i = blockIdx.x * blockDim.x + threadIdx.x;
    if (i < nE) atomicAdd(&deg[dst[i]], 1.0f);
}

__global__ void finish_deg_k(float* deg, int n) {
    int i = blockIdx.x * blockDim.x + threadIdx.x;
    if (i < n) {
        float d = deg[i] + 1.0f;          // self loop
        deg[i] = rsqrtf(d);               // deg >= 1 always
    }
}

// ---------------- GEMM: H = act(A) @ W  (A:[n,128], W:[128,128]) ----------------
// Block: 256 threads (8 waves). Each block computes a 16-row x 128-col tile.
// Wave w owns output columns [16w, 16w+16). K is chunked by KC through LDS.
// W chunk staged memory->LDS with async loads (no VGPR round-trip).

__global__ __launch_bounds__(256) void gemm_wmma_k(const float* __restrict__ A,
                                                   const float* __restrict__ W,
                                                   float* __restrict__ H,
                                                   int n, int leaky) {
    __shared__ float lw[KC * LDSP];   // 64x132 fp32 = 33792 B
    __shared__ float la[16 * LDSP];   // 16x132 fp32 =  8448 B

    const int tid  = threadIdx.x;
    const int base = blockIdx.x * 16;                // first node row of tile
    const unsigned lw_base = lds_offset_of(lw);

    // --- stage A tile (16 x 128) via VGPRs, LeakyReLU fused for layer 2 ---
    #pragma unroll
    for (int i = 0; i < 2; ++i) {
        int f4 = tid + i * 256;                      // 0..511 float4 slots
        int r  = f4 >> 5;                            // 0..15
        int c4 = (f4 & 31) << 2;                     // 0..124
        int row = base + r;
        if (row >= n) row = n - 1;                   // clamp (tail-safe)
        float4 v = *(const float4*)(A + (size_t)row * DF + c4);
        if (leaky) {
            v.x = v.x >= 0.0f ? v.x : 0.1f * v.x;
            v.y = v.y >= 0.0f ? v.y : 0.1f * v.y;
            v.z = v.z >= 0.0f ? v.z : 0.1f * v.z;
            v.w = v.w >= 0.0f ? v.w : 0.1f * v.w;
        }
        *(float4*)(la + r * LDSP + c4) = v;
    }

    const int wave = tid >> 5;
    const int lane = tid & 31;
    const int n0   = wave * 16;                      // output col base of this wave
    const int m    = lane & 15;                      // row (A) / col (B,C) within tile
    const int koff = (lane >> 4) * 2;                // K sub-offset per WMMA layout
    const int rowadd = (lane >> 4) * 8;              // C/D: VGPR j -> rows j, j+8

    v8f c = {};                                      // 16x16 f32 accumulator

    for (int kc = 0; kc < DF; kc += KC) {
        __syncthreads();                             // lw safe to overwrite / la ready
        // --- async stage W rows [kc, kc+KC): 64x128 -> 2048 b128 xfers, 8/thread ---
        #pragma unroll
        for (int i = 0; i < 8; ++i) {
            int f4 = tid + i * 256;                  // 0..2047
            int r  = f4 >> 5;                        // 0..63
            int c4 = (f4 & 31) << 2;                 // 0..124
            unsigned long long gaddr =
                (unsigned long long)(W + (size_t)(kc + r) * DF + c4);
            unsigned laddr = lw_base + (unsigned)(r * LDSP + c4) * 4u;
            asm volatile("global_load_async_to_lds_b128 %0, %1, off"
                         :: "v"(laddr), "v"(gaddr) : "memory");
        }
        asm volatile("s_wait_asynccnt 0x0" ::: "memory");
        __syncthreads();

        #pragma unroll
        for (int k = 0; k < KC; k += 4) {
            // A fragment: 16x4 f32 -> lane holds A[m][k+koff], A[m][k+koff+1]
            v2f a;
            a.x = la[m * LDSP + kc + k + koff];
            a.y = la[m * LDSP + kc + k + koff + 1];
            // B fragment: 4x16 f32 -> lane holds W[k+koff][n0+m], W[k+koff+1][n0+m]
            v2f b;
            b.x = lw[(k + koff) * LDSP + n0 + m];
            b.y = lw[(k + koff + 1) * LDSP + n0 + m];
            c = __builtin_amdgcn_wmma_f32_16x16x4_f32(false, a, false, b,
                                                      (short)0, c, false, false);
        }
    }

    // --- store C: VGPR j -> row (j + rowadd), col (n0 + m) ---
    float* dstp = H + (size_t)base * DF + n0 + m;
    #pragma unroll
    for (int j = 0; j < 8; ++j) {
        int row = j + rowadd;
        if (base + row < n)
            dstp[(size_t)row * DF] = c[j];
    }
}

// ---------------- self-loop + bias seed: out[v] = dinv[v]^2 * h[v] + b ----------------

__global__ void init_out_k(const float* __restrict__ h, const float* __restrict__ dinv,
                           const float* __restrict__ b, float* __restrict__ out, int n) {
    int idx = blockIdx.x * blockDim.x + threadIdx.x;   // one thread per (node, float4)
    if (idx >= n * 32) return;
    int v  = idx >> 5;
    int c4 = idx & 31;
    float di = dinv[v];
    float w  = di * di;
    float4 hv = ((const float4*)(h + (size_t)v * DF))[c4];
    float4 bv = ((const float4*)b)[c4];
    float4 o;
    o.x = hv.x * w + bv.x;
    o.y = hv.y * w + bv.y;
    o.z = hv.z * w + bv.z;
    o.w = hv.w * w + bv.w;
    ((float4*)(out + (size_t)v * DF))[c4] = o;
}

// ---------------- edge aggregation: wave per edge ----------------

__global__ __launch_bounds__(256) void edge_agg_k(const float* __restrict__ h,
                                                  const float* __restrict__ dinv,
                                                  const int* __restrict__ src,
                                                  const int* __restrict__ dst,
                                                  float* out, int nE) {
    int wid  = (blockIdx.x * blockDim.x + threadIdx.x) >> 5;
    int lane = threadIdx.x & 31;
    if (wid >= nE) return;
    int s = src[wid];
    int d = dst[wid];
    float nrm = dinv[s] * dinv[d];
    float4 v = ((const float4*)(h + (size_t)s * DF))[lane];   // coalesced 512B/wave
    float* o = out + (size_t)d * DF + (lane << 2);
    atomicAdd(o + 0, v.x * nrm);
    atomicAdd(o + 1, v.y * nrm);
    atomicAdd(o + 2, v.z * nrm);
    atomicAdd(o + 3, v.w * nrm);
}

// ---------------- final feature mean: wave per node ----------------

__global__ void mean_k(const float* __restrict__ out128, float* __restrict__ res, int n) {
    int wid  = (blockIdx.x * blockDim.x + threadIdx.x) >> 5;
    int lane = threadIdx.x & 31;
    if (wid >= n) return;
    float4 v = ((const float4*)(out128 + (size_t)wid * DF))[lane];
    float s = v.x + v.y + v.z + v.w;
    #pragma unroll
    for (int off = 16; off > 0; off >>= 1)
        s += __shfl_xor(s, off, 32);
    if (lane == 0) res[wid] = s * (1.0f / 128.0f);
}

// ---------------- launcher ----------------

extern "C" void kernel_launch(void* const* d_in, const int* in_sizes, int n_in,
                              void* d_out, int out_size, void* d_ws, size_t ws_size,
                              hipStream_t stream) {
    const float* x  = (const float*)d_in[0];
    const int*   ei = (const int*)d_in[1];
    const float* W1 = (const float*)d_in[2];
    const float* b1 = (const float*)d_in[3];
    const float* W2 = (const float*)d_in[4];
    const float* b2 = (const float*)d_in[5];

    const int N = in_sizes[0] / DF;       // 50000
    const int E = in_sizes[1] / 2;        // 1600000
    const int* src = ei;
    const int* dst = ei + E;

    // workspace: dinv[N] | h[N*128] | out[N*128]   (~51.4 MB total)
    float* dinv = (float*)d_ws;
    size_t npad = ((size_t)N + 255) & ~(size_t)255;
    float* h    = dinv + npad;
    float* o    = h + (size_t)N * DF;
    float* res  = (float*)d_out;

    const int TPB = 256;
    int gemm_blocks = (N + 15) / 16;

    // degrees -> dinv
    zero_k      <<<(N + TPB - 1) / TPB, TPB, 0, stream>>>(dinv, N);
    count_deg_k <<<(E + TPB - 1) / TPB, TPB, 0, stream>>>(dst, dinv, E);
    finish_deg_k<<<(N + TPB - 1) / TPB, TPB, 0, stream>>>(dinv, N);

    // layer 1
    gemm_wmma_k <<<gemm_blocks, TPB, 0, stream>>>(x, W1, h, N, /*leaky=*/0);
    init_out_k  <<<(N * 32 + TPB - 1) / TPB, TPB, 0, stream>>>(h, dinv, b1, o, N);
    edge_agg_k  <<<(E + 7) / 8, TPB, 0, stream>>>(h, dinv, src, dst, o, E);

    // layer 2 (LeakyReLU fused into GEMM A-load)
    gemm_wmma_k <<<gemm_blocks, TPB, 0, stream>>>(o, W2, h, N, /*leaky=*/1);
    init_out_k  <<<(N * 32 + TPB - 1) / TPB, TPB, 0, stream>>>(h, dinv, b2, o, N);
    edge_agg_k  <<<(E + 7) / 8, TPB, 0, stream>>>(h, dinv, src, dst, o, E);

    // mean over features
    mean_k      <<<(N + 7) / 8, TPB, 0, stream>>>(o, res, N);
}